// Model_51453708206399
// MI455X (gfx1250) — compile-verified
//
#include <hip/hip_runtime.h>
#include <hip/hip_bf16.h>
#include <stdint.h>

// Problem constants (match reference): B=32, S=8192, U=1024, H=1024
#define BB 32
#define SS 8192
#define UU 1024
#define HH 1024

typedef int      v4i __attribute__((ext_vector_type(4)));
typedef float    v4f __attribute__((ext_vector_type(4)));
typedef _Float16 v4h __attribute__((ext_vector_type(4)));

// One block per (b, s) row. 256 threads x 4 elements = 1024 = H.
// Copy rows (87.5%): nontemporal b128 load (int32-encoded int8) -> cvt -> NT b128 store.
// Update rows (12.5%): async-stage the lane's 8 bytes of the f16 update row into LDS
//   (gfx1250 global_load_async_to_lds_b64, tracked by ASYNCcnt), wait, read back,
//   quantize q = clip(rint(u/scale + zp), -128, 127) with +/-inf pinning, store float.
//   Each lane consumes exactly the bytes it staged, so no cross-wave barrier is needed:
//   s_wait_asynccnt 0 alone makes the wave's LDS writes visible to itself.
__global__ __launch_bounds__(256)
void scatter_quant_kernel(const int* __restrict__ in,        // [B,S,H] int8 values in int32
                          const int* __restrict__ indices,   // [B]
                          const _Float16* __restrict__ upd,  // [B,U,H]
                          const float* __restrict__ scales,  // [H]
                          const float* __restrict__ zps,     // [H]
                          float* __restrict__ out)           // [B,S,H]
{
    const int row = blockIdx.x;            // 0 .. B*S-1
    const int b   = row >> 13;             // row / S   (S = 8192)
    const int s   = row & (SS - 1);        // row % S
    const int col = threadIdx.x << 2;      // 4 elements per thread
    const size_t base = (size_t)row * HH + col;   // 64-bit: buffers are 1 GiB

    const int start  = indices[b];         // uniform per block -> scalar load
    const unsigned u = (unsigned)(s - start);

    if (u < (unsigned)UU) {
        // ---- quantize path ----
        __shared__ __align__(16) _Float16 srow[HH];   // 2 KiB staging buffer

        const _Float16* gsrc = upd + ((size_t)b * UU + u) * HH + col;
        unsigned lds_off = (unsigned)(uintptr_t)&srow[col];  // low 32 bits of generic
                                                             // LDS ptr == LDS byte offset
        // gfx1250 async global->LDS copy, 8 bytes/lane (EXEC all-1s here:
        // block-uniform branch, full 256-thread blocks).
        asm volatile("global_load_async_to_lds_b64 %0, %1, off"
                     :: "v"(lds_off), "v"(gsrc)
                     : "memory");
        asm volatile("s_wait_asynccnt 0" ::: "memory");  // this wave's copy complete

        const v4h hv  = *reinterpret_cast<const v4h*>(&srow[col]);   // ds_load_b64
        const v4f scv = *reinterpret_cast<const v4f*>(scales + col); // hot in L2 (4 KiB)
        const v4f zpv = *reinterpret_cast<const v4f*>(zps + col);

        v4f o;
#pragma unroll
        for (int k = 0; k < 4; ++k) {
            const float x = (float)hv[k];          // f16 -> f32 exact
            float q = rintf(x / scv[k] + zpv[k]);  // exact f32 divide, RNE rounding
            q = fmaxf(q, -128.0f);
            q = fminf(q,  127.0f);
            q = (x ==  (float)INFINITY) ?  127.0f : q;
            q = (x == -(float)INFINITY) ? -128.0f : q;
            o[k] = q;   // integer-valued in [-128,127] == (float)(int8)q
        }
        __builtin_nontemporal_store(o, reinterpret_cast<v4f*>(out + base));
    } else {
        // ---- copy path: pure bandwidth, keep the 2 GiB stream out of L2 ----
        const v4i v = __builtin_nontemporal_load(reinterpret_cast<const v4i*>(in + base));
        v4f o;
        o.x = (float)v.x;
        o.y = (float)v.y;
        o.z = (float)v.z;
        o.w = (float)v.w;
        __builtin_nontemporal_store(o, reinterpret_cast<v4f*>(out + base));
    }
}

extern "C" void kernel_launch(void* const* d_in, const int* in_sizes, int n_in,
                              void* d_out, int out_size, void* d_ws, size_t ws_size,
                              hipStream_t stream) {
    // setup_inputs order: input(int8->int32), indices(int32), updates(f16),
    //                     quant_scales(f32), quant_zero_points(f32), axis(int, ==1)
    const int*      in      = (const int*)d_in[0];
    const int*      indices = (const int*)d_in[1];
    const _Float16* upd     = (const _Float16*)d_in[2];
    const float*    scales  = (const float*)d_in[3];
    const float*    zps     = (const float*)d_in[4];
    float*          out     = (float*)d_out;

    const int num_rows = BB * SS;          // 262144 blocks, one row each
    scatter_quant_kernel<<<num_rows, 256, 0, stream>>>(in, indices, upd, scales, zps, out);
}